// MaskedConv2d_18708877541590
// MI455X (gfx1250) — compile-verified
//
#include <hip/hip_runtime.h>

typedef float v2f __attribute__((ext_vector_type(2)));
typedef float v8f __attribute__((ext_vector_type(8)));

namespace {
constexpr int CIN  = 32;
constexpr int COUT = 32;
constexpr int HH   = 256;
constexpr int WW   = 256;
constexpr int KTOT = CIN * 9;          // 288, permuted order: k' = r*32 + c
constexpr int HW   = HH * WW;          // 65536
}

// Implicit-GEMM masked 3x3 conv with native fp32 WMMA.
// K permuted tap-major (k' = r*32 + c) so each K=4 block keeps one (dh,dw):
// B-operand loads become fixed-stride immediate-offset loads from one base.
// Each wave: M = 32 (both channel tiles) x N = 32 (two 16-wide spatial tiles),
// one workgroup = one full (img, h) output row of 256 pixels.
__global__ __launch_bounds__(256) void masked_conv2d_wmma_f32(
    const float* __restrict__ x,     // (16, 32, 256, 256)
    const float* __restrict__ w,     // (32, 32, 3, 3) -> [o][j], j = c*9 + r
    const float* __restrict__ thrp,  // (1,)
    float* __restrict__ out)         // (16, 32, 256, 256)
{
    // sW[(k'>>2)*128 + o*4 + (k'&3)] : per-lane A fragment is one b64,
    // lanes 0-15 banks {4L,4L+1}, lanes 16-31 {4L+2,4L+3} -> conflict-free.
    __shared__ float sW[(KTOT / 4) * COUT * 4]; // 36 KB
    __shared__ float sScore[KTOT];
    __shared__ int   sCount;

    const int tid = threadIdx.x;
    const float thr = fmaxf(thrp[0], 0.0f);

    if (tid == 0) sCount = 0;
    __syncthreads();

    // ---- Phase A: row score = mean_o |w[o][j]|; count surviving rows ----
    for (int j = tid; j < KTOT; j += 256) {
        float s = 0.0f;
        #pragma unroll
        for (int o = 0; o < COUT; ++o) s += fabsf(w[o * KTOT + j]);
        s *= (1.0f / (float)COUT);
        sScore[j] = s;
        if (s - thr > 0.0f) atomicAdd(&sCount, 1);
    }
    __syncthreads();

    // ratio <= 0.01 with 288 rows <=> count <= 2 -> fallback mask (score > 0)
    const bool fallback = (sCount <= 2);

    // ---- Phase B: masked weights into LDS, K permuted tap-major ----
    for (int idx = tid; idx < KTOT * COUT; idx += 256) {
        const int kp = idx >> 5;          // permuted k' = r*32 + c
        const int o  = idx & 31;
        const int r  = kp >> 5;           // tap 0..8
        const int c  = kp & 31;           // input channel
        const int j  = c * 9 + r;         // original weight row index
        const float s = sScore[j];
        const float keep = fallback ? (s > 0.0f ? 1.0f : 0.0f)
                                    : ((s - thr) > 0.0f ? 1.0f : 0.0f);
        sW[((kp >> 2) << 7) + (o << 2) + (kp & 3)] = w[o * KTOT + j] * keep;
    }
    __syncthreads();

    // ---- Phase C: implicit GEMM ----
    const int wave   = tid >> 5;
    const int lane   = tid & 31;
    const int lane16 = lane & 15;
    const int qh     = (lane >> 4) << 1;   // K sub-row 0 (lo half) / 2 (hi half)

    const int row = blockIdx.x;            // one (img, h) output row per block
    const int img = row >> 8;
    const int h   = row & (HH - 1);
    const int w0  = wave * 32;             // this wave's 32 spatial columns

    const float* __restrict__ xi = x + (size_t)img * CIN * HW;

    v8f acc00 = {};  // M tile 0 (o 0..15)  x N tile 0
    v8f acc10 = {};  // M tile 1 (o 16..31) x N tile 0
    v8f acc01 = {};  // M tile 0            x N tile 1
    v8f acc11 = {};  // M tile 1            x N tile 1

    const int aOff0 = (lane16 << 2) + qh;          // LDS float offset, M tile 0
    const int aOff1 = ((lane16 + 16) << 2) + qh;   // M tile 1

    for (int r = 0; r < 9; ++r) {
        const int dh = (r * 11) >> 5;              // r/3 for r in [0,8]
        const int dw = r - dh * 3;
        const int hv = h + dh - 1;
        const bool okh = (unsigned)hv < (unsigned)HH;
        const int hc = min(max(hv, 0), HH - 1);

        // per-lane clamped base pointers for both N tiles (include qh channel)
        const int wv0 = w0 + lane16 + dw - 1;
        const bool ok0 = okh & ((unsigned)wv0 < (unsigned)WW);
        const int wc0 = min(max(wv0, 0), WW - 1);
        const float* __restrict__ bp0 = xi + (size_t)qh * HW + hc * WW + wc0;

        const int wv1 = wv0 + 16;
        const bool ok1 = okh & ((unsigned)wv1 < (unsigned)WW);
        const int wc1 = min(max(wv1, 0), WW - 1);
        const float* __restrict__ bp1 = xi + (size_t)qh * HW + hc * WW + wc1;

        const float* __restrict__ arow = &sW[r << 10];   // r * 8 * 128 floats

        #pragma unroll
        for (int cb = 0; cb < 32; cb += 4) {
            // B loads: fixed-stride immediate offsets (cb*HW*4 < 8 MB)
            const float b0a = bp0[cb * HW];
            const float b0b = bp0[(cb + 1) * HW];
            const float b1a = bp1[cb * HW];
            const float b1b = bp1[(cb + 1) * HW];
            v2f B0, B1;
            B0.x = ok0 ? b0a : 0.0f;  B0.y = ok0 ? b0b : 0.0f;
            B1.x = ok1 ? b1a : 0.0f;  B1.y = ok1 ? b1b : 0.0f;

            const int rb = (cb >> 2) << 7;
            const v2f A0 = *(const v2f*)&arow[rb + aOff0];
            const v2f A1 = *(const v2f*)&arow[rb + aOff1];

            acc00 = __builtin_amdgcn_wmma_f32_16x16x4_f32(
                        false, A0, false, B0, (short)0, acc00, false, false);
            acc10 = __builtin_amdgcn_wmma_f32_16x16x4_f32(
                        false, A1, false, B0, (short)0, acc10, false, false);
            acc01 = __builtin_amdgcn_wmma_f32_16x16x4_f32(
                        false, A0, false, B1, (short)0, acc01, false, false);
            acc11 = __builtin_amdgcn_wmma_f32_16x16x4_f32(
                        false, A1, false, B1, (short)0, acc11, false, false);
        }
    }

    // ---- Store: VGPR v -> M = v (lanes 0-15) / v+8 (lanes 16-31), N = lane&15
    const int mhi = (lane >> 4) << 3;  // 0 or 8
    float* __restrict__ op0 =
        out + (size_t)img * COUT * HW + (size_t)h * WW + w0 + lane16;
    float* __restrict__ op1 = op0 + 16;
    #pragma unroll
    for (int v = 0; v < 8; ++v) {
        const int m = v + mhi;
        op0[(size_t)m * HW]        = acc00[v];
        op0[(size_t)(m + 16) * HW] = acc10[v];
        op1[(size_t)m * HW]        = acc01[v];
        op1[(size_t)(m + 16) * HW] = acc11[v];
    }
}

extern "C" void kernel_launch(void* const* d_in, const int* in_sizes, int n_in,
                              void* d_out, int out_size, void* d_ws, size_t ws_size,
                              hipStream_t stream) {
    (void)in_sizes; (void)n_in; (void)d_ws; (void)ws_size; (void)out_size;
    const float* x   = (const float*)d_in[0];
    const float* w   = (const float*)d_in[1];
    const float* thr = (const float*)d_in[2];
    float* out       = (float*)d_out;

    dim3 grid(16 * HH);   // one (img, h) output row per block: 4096 blocks
    dim3 block(256);      // 8 wave32
    masked_conv2d_wmma_f32<<<grid, block, 0, stream>>>(x, w, thr, out);
}